// DynamicKGE_56023553409042
// MI455X (gfx1250) — compile-verified
//
#include <hip/hip_runtime.h>
#include <hip/hip_bf16.h>

#define N_ENT   50000
#define N_RELS  8
#define RTYPES  16      // 2 * N_RELS
#define DIM     128
#define N_EDGES 800000

typedef __attribute__((ext_vector_type(2))) float v2f;
typedef __attribute__((ext_vector_type(8))) float v8f;

// ---------------------------------------------------------------------------
// Kernel 1: sigmoid gates + relation_o (tiny: 8x128). Replicates the source
// bug: relation_o = sig(gate_rel)*rel_emb + (1 - sig(gate_ent))*rel_ctx.
// ---------------------------------------------------------------------------
__global__ void gates_rel_kernel(const float* __restrict__ gate_e,
                                 const float* __restrict__ gate_r,
                                 const float* __restrict__ rel_emb,
                                 const float* __restrict__ rel_ctx,
                                 float* __restrict__ sig_ge,
                                 float* __restrict__ sig_gr,
                                 float* __restrict__ rel_out) {
  int t = blockIdx.x * blockDim.x + threadIdx.x;
  if (t < DIM) {
    sig_ge[t] = 1.f / (1.f + __expf(-gate_e[t]));
    sig_gr[t] = 1.f / (1.f + __expf(-gate_r[t]));
  }
  if (t < N_RELS * DIM) {
    int f = t & (DIM - 1);
    float ge = 1.f / (1.f + __expf(-gate_e[f]));
    float gr = 1.f / (1.f + __expf(-gate_r[f]));
    rel_out[t] = gr * rel_emb[t] + (1.f - ge) * rel_ctx[t];
  }
}

// ---------------------------------------------------------------------------
// Kernel 2: segment-sum scatter.
//   m[(dst*16 + type), d] += edge_norm[e] * entity_ctx[entity[src]][d]
// Thread t -> (edge e = t>>7, dim d = t&127): 512B-coalesced gather + atomic.
// ---------------------------------------------------------------------------
__global__ void __launch_bounds__(256)
scatter_kernel(const int* __restrict__ entity,
               const int* __restrict__ edge_index,   // [2, E] flat: row0=dst, row1=src
               const int* __restrict__ edge_type,
               const float* __restrict__ edge_norm,
               const float* __restrict__ entity_ctx,
               float* __restrict__ m) {
  long long t = (long long)blockIdx.x * blockDim.x + threadIdx.x;
  const long long total = (long long)N_EDGES * DIM;
  if (t >= total) return;
  int e = (int)(t >> 7);
  int d = (int)(t & (DIM - 1));
  int dst = edge_index[e];
  int src = edge_index[N_EDGES + e];
  int typ = edge_type[e];
  float v = entity_ctx[(long long)entity[src] * DIM + d] * edge_norm[e];
  atomicAdd(&m[((long long)dst * RTYPES + typ) * DIM + d], v);
}

// ---------------------------------------------------------------------------
// Kernel 3: grouped GEMM via V_WMMA_F32_16X16X4_F32 + fused ReLU/gate epilogue.
// One wave = 16 nodes x 128 features (8 accumulators of v8f).
// K = 17 segments of 128 (16 relation slices of m, then x @ root).
// B panel (128x128) staged in LDS column-major (stride 130) by the whole block.
// ---------------------------------------------------------------------------
__global__ void __launch_bounds__(256)
rgcn_wmma_kernel(const int* __restrict__ entity,
                 const float* __restrict__ m,
                 const float* __restrict__ entity_ctx,
                 const float* __restrict__ Wrel,     // [16,128,128]
                 const float* __restrict__ root,     // [128,128]
                 const float* __restrict__ entity_emb,
                 const float* __restrict__ sig_ge,
                 float* __restrict__ out) {
  const int LDB = DIM + 2;                 // pad: conflict-free, keeps 8B align
  __shared__ float ldsB[DIM * LDB];        // ~66.5 KB

  int lane = threadIdx.x & 31;
  int wave = blockIdx.x * (blockDim.x >> 5) + (threadIdx.x >> 5);
  const int NT = N_ENT / 16;               // 3125 tiles (exact)
  bool active = wave < NT;                 // wave-uniform: EXEC all-1s inside
  int rowBase = active ? wave * 16 : 0;

  int halfk = lane >> 4;                   // 0: K=k,k+1   1: K=k+2,k+3
  int lm = lane & 15;                      // A row / B col within tile

  const float* mrow = m + (long long)(rowBase + lm) * (RTYPES * DIM);
  const float* xrow = entity_ctx + (long long)entity[rowBase + lm] * DIM;

  v8f acc[8];
#pragma unroll
  for (int i = 0; i < 8; ++i)
    acc[i] = (v8f){0.f, 0.f, 0.f, 0.f, 0.f, 0.f, 0.f, 0.f};

  for (int seg = 0; seg <= RTYPES; ++seg) {
    const float* Bmat = (seg < RTYPES) ? (Wrel + (size_t)seg * DIM * DIM) : root;

    __syncthreads();
    // stage B column-major: ldsB[col*LDB + k] = B[k, col]
    for (int i = threadIdx.x; i < DIM * DIM; i += blockDim.x) {
      int kI = i >> 7;
      int col = i & (DIM - 1);
      ldsB[col * LDB + kI] = Bmat[i];
    }
    __syncthreads();

    if (active) {
      const float* Arow = (seg < RTYPES) ? (mrow + seg * DIM) : xrow;
#pragma unroll 4
      for (int k = 0; k < DIM; k += 4) {
        int kk = k + 2 * halfk;
        v2f a;
        a.x = Arow[kk];                    // global_load_b64 (contiguous pair)
        a.y = Arow[kk + 1];
#pragma unroll
        for (int ft = 0; ft < 8; ++ft) {
          int col = ft * 16 + lm;
          v2f b = *(const v2f*)&ldsB[col * LDB + kk];   // ds_load_b64
          acc[ft] = __builtin_amdgcn_wmma_f32_16x16x4_f32(
              false, a, false, b, (short)0, acc[ft], false, false);
        }
      }
    }
  }

  if (active) {
    // C/D layout: element (vgpr v, lane L) -> row v + 8*(L>>4), col L&15
#pragma unroll
    for (int ft = 0; ft < 8; ++ft) {
      int f = ft * 16 + lm;
      float ge = sig_ge[f];
#pragma unroll
      for (int v = 0; v < 8; ++v) {
        int node = rowBase + v + 8 * halfk;
        float c = acc[ft][v];
        float r = c > 0.f ? c : 0.f;       // ReLU
        float ee = entity_emb[(size_t)entity[node] * DIM + f];
        out[(size_t)node * DIM + f] = ge * ee + (1.f - ge) * r;
      }
    }
  }
}

// ---------------------------------------------------------------------------
extern "C" void kernel_launch(void* const* d_in, const int* in_sizes, int n_in,
                              void* d_out, int out_size, void* d_ws, size_t ws_size,
                              hipStream_t stream) {
  const int*   entity     = (const int*)d_in[0];
  const int*   edge_index = (const int*)d_in[1];
  const int*   edge_type  = (const int*)d_in[2];
  const float* edge_norm  = (const float*)d_in[3];
  /* d_in[4] = DAD_rel: unused by the reference forward pass */
  const float* entity_emb = (const float*)d_in[5];
  const float* rel_emb    = (const float*)d_in[6];
  const float* entity_ctx = (const float*)d_in[7];
  const float* rel_ctx    = (const float*)d_in[8];
  const float* gate_e     = (const float*)d_in[9];
  const float* gate_r     = (const float*)d_in[10];
  const float* Wrel       = (const float*)d_in[11];
  const float* root       = (const float*)d_in[12];

  float* out_entity = (float*)d_out;                       // [50000,128]
  float* out_rel    = out_entity + (size_t)N_ENT * DIM;    // [8,128]

  // workspace: m [N*16*128] f32 (~409.6 MB) + sigmoid gates
  float* m = (float*)d_ws;
  const size_t m_elems = (size_t)N_ENT * RTYPES * DIM;
  float* sig_ge = m + m_elems;
  float* sig_gr = sig_ge + DIM;

  hipMemsetAsync(m, 0, m_elems * sizeof(float), stream);

  gates_rel_kernel<<<(N_RELS * DIM + 255) / 256, 256, 0, stream>>>(
      gate_e, gate_r, rel_emb, rel_ctx, sig_ge, sig_gr, out_rel);

  long long total = (long long)N_EDGES * DIM;
  scatter_kernel<<<(int)((total + 255) / 256), 256, 0, stream>>>(
      entity, edge_index, edge_type, edge_norm, entity_ctx, m);

  int tiles = N_ENT / 16;                 // 3125
  int blocks = (tiles + 7) / 8;           // 8 waves / block
  rgcn_wmma_kernel<<<blocks, 256, 0, stream>>>(
      entity, m, entity_ctx, Wrel, root, entity_emb, sig_ge, out_entity);
}